// StudSarNeural_59742995087659
// MI455X (gfx1250) — compile-verified
//
#include <hip/hip_runtime.h>
#include <hip/hip_bf16.h>
#include <cmath>

typedef float v8f __attribute__((ext_vector_type(8)));
typedef float v2f __attribute__((ext_vector_type(2)));

#define HAVE_WMMA_F32_4 __has_builtin(__builtin_amdgcn_wmma_f32_16x16x4_f32)

// ---------------------------------------------------------------------------
// Kernel 1: fused scoring.  One pass over the 1 GiB embedding matrix.
// Block = 256 threads = 8 waves; each wave owns 16 rows (128 rows / block).
// Per wave: WMMA f32 16x16x4 chain computes dot(row, query) for its 16 rows;
// VALU fuses the per-row sum-of-squares in the same streamed loads.
// ---------------------------------------------------------------------------
__global__ void __launch_bounds__(256)
score_kernel(const float* __restrict__ q,
             const float* __restrict__ mem,
             const float* __restrict__ rep,
             float* __restrict__ scored) {
  __shared__ float s_q[256] __attribute__((aligned(16)));
  __shared__ float s_red[256];
  __shared__ float s_ssq[128];
#if !HAVE_WMMA_F32_4
  __shared__ float s_dot[128];
#endif
  __shared__ float s_qn;

  const int t = threadIdx.x;

  // Load query to LDS and compute ||q|| with a deterministic tree reduction.
  float qv = q[t];
  s_q[t] = qv;
  s_red[t] = qv * qv;
  __syncthreads();
  for (int off = 128; off > 0; off >>= 1) {
    if (t < off) s_red[t] += s_red[t + off];
    __syncthreads();
  }
  if (t == 0) s_qn = fmaxf(sqrtf(s_red[0]), 1e-8f);
  __syncthreads();

  const int wave = t >> 5;
  const int lane = t & 31;
  const int half = lane >> 4;          // K-half of the wave (WMMA f32 A layout)
  const int r    = lane & 15;          // row within the wave's 16-row tile
  const long long rowBase = (long long)blockIdx.x * 128 + (long long)wave * 16;
  const long long row = rowBase + r;
  const float* rowp = mem + row * 256 + half * 2;

  float ssq = 0.0f;
#if HAVE_WMMA_F32_4
  v8f c = {0.f, 0.f, 0.f, 0.f, 0.f, 0.f, 0.f, 0.f};
#pragma unroll 4
  for (int k = 0; k < 256; k += 4) {
    // A: lane L<16 holds row L, K={k,k+1}; lane>=16 holds K={k+2,k+3}.
    v2f a = __builtin_nontemporal_load((const v2f*)(rowp + k));
    // B: query chunk, replicated across all 16 N columns.
    v2f b = *(const v2f*)(&s_q[k + half * 2]);
    ssq = fmaf(a[0], a[0], ssq);
    ssq = fmaf(a[1], a[1], ssq);
    c = __builtin_amdgcn_wmma_f32_16x16x4_f32(
        /*neg_a=*/false, a, /*neg_b=*/false, b,
        /*c_mod=*/(short)0, c, /*reuse_a=*/false, /*reuse_b=*/false);
  }
#else
  float dotp = 0.0f;
#pragma unroll 4
  for (int k = 0; k < 256; k += 4) {
    v2f a = __builtin_nontemporal_load((const v2f*)(rowp + k));
    v2f b = *(const v2f*)(&s_q[k + half * 2]);
    ssq  = fmaf(a[0], a[0], ssq);
    ssq  = fmaf(a[1], a[1], ssq);
    dotp = fmaf(a[0], b[0], dotp);
    dotp = fmaf(a[1], b[1], dotp);
  }
#endif

  // Combine the two K-halves of the per-row sum of squares (wave32 xor-16).
  float ssq_full = ssq + __shfl_xor(ssq, 16);
  if (half == 0) s_ssq[wave * 16 + r] = ssq_full;
#if !HAVE_WMMA_F32_4
  float dot_full = dotp + __shfl_xor(dotp, 16);
  if (half == 0) s_dot[wave * 16 + r] = dot_full;
#endif
  __syncthreads();

  // C layout: lane 0 holds rows 0..7 (VGPR 0..7), lane 16 holds rows 8..15.
  if (r == 0) {
    const float qn = s_qn;
#pragma unroll
    for (int j = 0; j < 8; ++j) {
      const long long m = rowBase + half * 8 + j;
#if HAVE_WMMA_F32_4
      const float dot = c[j];
#else
      const float dot = s_dot[wave * 16 + half * 8 + j];
#endif
      const float ss   = s_ssq[wave * 16 + half * 8 + j];
      const float en   = fmaxf(sqrtf(ss), 1e-8f);
      const float sim  = dot / (en * qn);
      const float boost = 1.0f + 0.1f * tanhf(0.5f * rep[m]);
      scored[m] = sim * boost;
    }
  }
}

// ---------------------------------------------------------------------------
// Bitonic descending sort of 1024 (val, idx) pairs in LDS, 256 threads.
// ---------------------------------------------------------------------------
__device__ __forceinline__ void bitonic_sort_desc_1024(float* v, int* idx) {
  for (int klen = 2; klen <= 1024; klen <<= 1) {
    for (int j = klen >> 1; j > 0; j >>= 1) {
      __syncthreads();
      for (int e = threadIdx.x; e < 1024; e += 256) {
        const int p = e ^ j;
        if (p > e) {
          const float a = v[e], b = v[p];
          const bool desc = (e & klen) == 0;
          const bool sw = desc ? (a < b) : (a > b);
          if (sw) {
            v[e] = b; v[p] = a;
            const int ti = idx[e]; idx[e] = idx[p]; idx[p] = ti;
          }
        }
      }
    }
  }
  __syncthreads();
}

__global__ void __launch_bounds__(256)
topk_stage(const float* __restrict__ in_v, const int* __restrict__ in_i, int n,
           float* __restrict__ out_v, int* __restrict__ out_i) {
  __shared__ float sv[1024];
  __shared__ int   si[1024];
  const int t = threadIdx.x;
  const int base = blockIdx.x * 1024;
  for (int e = t; e < 1024; e += 256) {
    const int g = base + e;
    const bool ok = g < n;
    sv[e] = ok ? in_v[g] : -INFINITY;
    si[e] = ok ? (in_i ? in_i[g] : g) : -1;
  }
  __syncthreads();
  bitonic_sort_desc_1024(sv, si);
  if (t < 64) {
    out_v[blockIdx.x * 64 + t] = sv[t];
    out_i[blockIdx.x * 64 + t] = si[t];
  }
}

__global__ void __launch_bounds__(256)
topk_final(const float* __restrict__ in_v, const int* __restrict__ in_i, int n,
           float* __restrict__ out) {
  __shared__ float sv[1024];
  __shared__ int   si[1024];
  const int t = threadIdx.x;
  for (int e = t; e < 1024; e += 256) {
    const bool ok = e < n;
    sv[e] = ok ? in_v[e] : -INFINITY;
    si[e] = ok ? in_i[e] : -1;
  }
  __syncthreads();
  bitonic_sort_desc_1024(sv, si);
  if (t < 64) {
    out[t]      = sv[t];          // top values, sorted descending
    out[64 + t] = (float)si[t];   // indices, emitted as floats
  }
}

// ---------------------------------------------------------------------------
extern "C" void kernel_launch(void* const* d_in, const int* in_sizes, int n_in,
                              void* d_out, int out_size, void* d_ws, size_t ws_size,
                              hipStream_t stream) {
  const float* q   = (const float*)d_in[0];   // [256]
  const float* mem = (const float*)d_in[1];   // [N,256]
  const float* rep = (const float*)d_in[2];   // [N]
  const int N = in_sizes[2];
  float* out = (float*)d_out;                 // 64 vals ++ 64 idx (as float)

  // Workspace carve-up (floats): scored[N], two ping-pong candidate buffers.
  float* scored = (float*)d_ws;
  const int cap = ((N + 1023) / 1024) * 64;   // max candidates after stage 1
  float* bv[2];
  int*   bi[2];
  bv[0] = scored + N;
  bi[0] = (int*)(bv[0] + cap);
  bv[1] = (float*)(bi[0] + cap);
  bi[1] = (int*)(bv[1] + cap);

  // Stage 0: fused scoring (one streamed pass over 1 GiB).
  const int nBlocks = N / 128;                // 128 rows per 256-thread block
  score_kernel<<<nBlocks, 256, 0, stream>>>(q, mem, rep, scored);

  // Top-k funnel: 1M -> 64K -> 4K -> 256 -> 64.
  const float* cur_v = scored;
  const int*   cur_i = nullptr;
  int n = N, pp = 0;
  while (n > 1024) {
    const int nb = (n + 1023) / 1024;
    topk_stage<<<nb, 256, 0, stream>>>(cur_v, cur_i, n, bv[pp], bi[pp]);
    cur_v = bv[pp];
    cur_i = bi[pp];
    n = nb * 64;
    pp ^= 1;
  }
  topk_final<<<1, 256, 0, stream>>>(cur_v, cur_i, n, out);
}